// CircularDND_12713103196904
// MI455X (gfx1250) — compile-verified
//
#include <hip/hip_runtime.h>

typedef __attribute__((ext_vector_type(2))) float v2f;
typedef __attribute__((ext_vector_type(8))) float v8f;

#define DELTA      0.001f
#define QW         50
#define MAXLEN     1000000
#define NCH        64
#define TILES      (MAXLEN / 16)                        /* 62500 */
#define K1_BLOCKS  512
#define K2_CHUNK   8192
#define K2_BLOCKS  ((MAXLEN + K2_CHUNK - 1) / K2_CHUNK) /* 123 */
#define NCAND      (K2_BLOCKS * QW)                     /* 6150 */
#define NEG_INF    (-3.402823466e38f)

// One async batch = 8 x b128 per wave = one 16x64 f32 tile (4 KB, contiguous).
// INST_OFFSET applies to BOTH the global and LDS addresses, so one address
// pair covers the whole tile.
#define ASYNC_LD(OFF)                                                      \
  asm volatile("global_load_async_to_lds_b128 %0, %1, off offset:" #OFF    \
               :: "v"(ldsA), "v"(gA) : "memory")

__device__ __forceinline__ void issue_tile_async(const float* keys, int tile,
                                                 unsigned ldsBase, int lane) {
  unsigned long long gA =
      (unsigned long long)(size_t)keys + (unsigned long long)tile * 4096ull +
      (unsigned)lane * 16u;
  unsigned ldsA = ldsBase + (unsigned)lane * 16u;
  ASYNC_LD(0);    ASYNC_LD(512);  ASYNC_LD(1024); ASYNC_LD(1536);
  ASYNC_LD(2048); ASYNC_LD(2560); ASYNC_LD(3072); ASYNC_LD(3584);
}

// ---------------------------------------------------------------------------
// Kernel 1: stream keys (256 MB, bandwidth-bound), compute
//   w_i = 1 / (||q - k_i||^2 + delta)
// Pipeline: global -> LDS via async-to-LDS (double-buffered per wave, tracked
// by ASYNCcnt), LDS -> VGPR via ds_load, then 16x V_WMMA_F32_16X16X4_F32
// computing diag((K-q)(K-q)^T) per 16-row tile (A registers reused as
// B == A^T by f32 layout symmetry).
// ---------------------------------------------------------------------------
__global__ __launch_bounds__(256)
void k1_weights(const float* __restrict__ q, const float* __restrict__ keys,
                float* __restrict__ wts, float* __restrict__ psum) {
  __shared__ __align__(16) float stage[8 * 2 * 1024];  // 8 waves x 2 x 4 KB
  __shared__ float ssum[256];

  const int tid   = threadIdx.x;
  const int lane  = tid & 31;
  const int half  = lane >> 4;          // 0: holds K0,K1   1: holds K2,K3
  const int mrow  = lane & 15;          // matrix row within tile
  const int wslot = tid >> 5;           // wave within block
  const int waveId = (blockIdx.x * blockDim.x + tid) >> 5;
  const int nWaves = (gridDim.x * blockDim.x) >> 5;

  const int      wbase   = wslot * 2048;               // element base of wave's LDS
  const unsigned ldsWave = (unsigned)(size_t)&stage[wbase];
  const int      myoff   = mrow * 64 + half * 2;       // lane's element offset

  // Preload the query components this lane needs for every K=4 chunk.
  v2f qreg[16];
#pragma unroll
  for (int kk = 0; kk < 16; ++kk)
    qreg[kk] = *(const v2f*)(q + kk * 4 + half * 2);

  // C layout: VGPR v, lanes0-15 -> M=v, lanes16-31 -> M=v+8.
  // diag(i<8) @ lane i vgpr i; diag(i>=8) @ lane i+16 vgpr i-8.
  const bool active = (lane < 8) || (lane >= 24);
  const int  rsel   = (lane < 8) ? lane : (lane - 16);

  float sumAcc = 0.f;
  int buf = 0;

  if (waveId < TILES) issue_tile_async(keys, waveId, ldsWave, lane);

  for (int t = waveId; t < TILES; t += nWaves) {
    const int tn = t + nWaves;
    if (tn < TILES) {
      issue_tile_async(keys, tn, ldsWave + (unsigned)((buf ^ 1) * 4096), lane);
      asm volatile("s_wait_asynccnt 0x8" ::: "memory");  // tile t complete
    } else {
      asm volatile("s_wait_asynccnt 0x0" ::: "memory");
    }

    // LDS -> registers: direct indexing of the __shared__ array so clang
    // keeps addrspace(3) and emits ds_load_b64 (not flat_load).
    const int eb = wbase + buf * 1024 + myoff;
    v2f a[16];
#pragma unroll
    for (int kk = 0; kk < 16; ++kk)
      a[kk] = *(const v2f*)&stage[eb + kk * 4];

    // D += A' * A'^T  (B operand = same registers; f32 B 4x16 layout is the
    // transposed mapping of the f32 A 16x4 layout). Two accumulators to
    // halve the WMMA dependency chain.
    v8f c0 = {}, c1 = {};
#pragma unroll
    for (int kk = 0; kk < 16; kk += 2) {
      v2f d0 = a[kk]     - qreg[kk];
      v2f d1 = a[kk + 1] - qreg[kk + 1];
      c0 = __builtin_amdgcn_wmma_f32_16x16x4_f32(false, d0, false, d0,
                                                 (short)0, c0, false, false);
      c1 = __builtin_amdgcn_wmma_f32_16x16x4_f32(false, d1, false, d1,
                                                 (short)0, c1, false, false);
    }

    // Extract diagonal -> squared distance for this lane's row.
    float diag = 0.f;
#pragma unroll
    for (int v = 0; v < 8; ++v) {
      if (lane == v || lane == v + 24) diag = c0[v] + c1[v];
    }

    if (active) {
      float w = 1.0f / (diag + DELTA);
      wts[t * 16 + rsel] = w;
      sumAcc += w;
    }
    buf ^= 1;
  }

  // Deterministic block reduction of the weight sum.
  ssum[tid] = sumAcc;
  __syncthreads();
  for (int s = 128; s > 0; s >>= 1) {
    if (tid < s) ssum[tid] += ssum[tid + s];
    __syncthreads();
  }
  if (tid == 0) psum[blockIdx.x] = ssum[0];
}

// ---------------------------------------------------------------------------
// Kernel 2: per-block top-50 over an 8192-element chunk of the weight array.
// Chunk cached in LDS; 50 rounds of block-wide argmax (min-index tiebreak).
// ---------------------------------------------------------------------------
__global__ __launch_bounds__(256)
void k2_blocktop(const float* __restrict__ wts,
                 float* __restrict__ candV, int* __restrict__ candI) {
  __shared__ float sv[K2_CHUNK];
  __shared__ float rv[256];
  __shared__ int   ri[256];
  const int tid  = threadIdx.x;
  const int base = blockIdx.x * K2_CHUNK;

  for (int i = tid; i < K2_CHUNK; i += 256) {
    int g = base + i;
    sv[i] = (g < MAXLEN) ? wts[g] : NEG_INF;
  }
  __syncthreads();

  for (int r = 0; r < QW; ++r) {
    float bv = NEG_INF; int bi = 0;
    for (int i = tid; i < K2_CHUNK; i += 256) {   // ascending -> min-idx on ties
      float v = sv[i];
      if (v > bv) { bv = v; bi = i; }
    }
    rv[tid] = bv; ri[tid] = bi;
    __syncthreads();
    for (int s = 128; s > 0; s >>= 1) {
      if (tid < s) {
        float ov = rv[tid + s]; int oi = ri[tid + s];
        if (ov > rv[tid] || (ov == rv[tid] && oi < ri[tid])) {
          rv[tid] = ov; ri[tid] = oi;
        }
      }
      __syncthreads();
    }
    if (tid == 0) {
      candV[blockIdx.x * QW + r] = rv[0];
      candI[blockIdx.x * QW + r] = base + ri[0];
      sv[ri[0]] = NEG_INF;
    }
    __syncthreads();
  }
}

// ---------------------------------------------------------------------------
// Kernel 3: single block. Merge 6150 candidates -> global top-50, sum the
// per-block partials in fixed order, gather value rows, weighted sum -> [64].
// ---------------------------------------------------------------------------
__global__ __launch_bounds__(256)
void k3_final(const float* __restrict__ candV, const int* __restrict__ candI,
              const float* __restrict__ psum, const float* __restrict__ values,
              float* __restrict__ out) {
  __shared__ float cv[NCAND];
  __shared__ int   ciL[NCAND];
  __shared__ float rv[256];
  __shared__ int   ri[256];
  __shared__ float topw[QW];
  __shared__ int   topi[QW];
  __shared__ float totalSh;
  const int tid = threadIdx.x;

  for (int i = tid; i < NCAND; i += 256) { cv[i] = candV[i]; ciL[i] = candI[i]; }
  if (tid == 0) {
    float tot = 0.f;
    for (int b = 0; b < K1_BLOCKS; ++b) tot += psum[b];   // fixed order
    totalSh = tot;
  }
  __syncthreads();

  for (int r = 0; r < QW; ++r) {
    float bv = NEG_INF; int bi = 0;
    for (int i = tid; i < NCAND; i += 256) {
      float v = cv[i];
      if (v > bv) { bv = v; bi = i; }
    }
    rv[tid] = bv; ri[tid] = bi;
    __syncthreads();
    for (int s = 128; s > 0; s >>= 1) {
      if (tid < s) {
        float ov = rv[tid + s]; int oi = ri[tid + s];
        if (ov > rv[tid] || (ov == rv[tid] && oi < ri[tid])) {
          rv[tid] = ov; ri[tid] = oi;
        }
      }
      __syncthreads();
    }
    if (tid == 0) {
      topw[r] = rv[0];
      topi[r] = ciL[ri[0]];
      cv[ri[0]] = NEG_INF;
    }
    __syncthreads();
  }

  if (tid < NCH) {
    float inv = 1.0f / totalSh;
    float acc = 0.f;
    for (int j = 0; j < QW; ++j)
      acc += (topw[j] * inv) * values[(size_t)topi[j] * NCH + tid];
    out[tid] = acc;
  }
}

// ---------------------------------------------------------------------------
extern "C" void kernel_launch(void* const* d_in, const int* in_sizes, int n_in,
                              void* d_out, int out_size, void* d_ws, size_t ws_size,
                              hipStream_t stream) {
  const float* q      = (const float*)d_in[0];   // [64]
  const float* keys   = (const float*)d_in[1];   // [1e6, 64]
  const float* values = (const float*)d_in[2];   // [1e6, 64]
  float* out = (float*)d_out;                    // [1, 64] -> 64 floats

  // Workspace carve-up (~4.05 MB): weights | partial sums | candidates
  float* wts   = (float*)d_ws;                   // MAXLEN floats
  float* psum  = wts + MAXLEN;                   // K1_BLOCKS floats
  float* candV = psum + K1_BLOCKS;               // NCAND floats
  int*   candI = (int*)(candV + NCAND);          // NCAND ints

  k1_weights<<<K1_BLOCKS, 256, 0, stream>>>(q, keys, wts, psum);
  k2_blocktop<<<K2_BLOCKS, 256, 0, stream>>>(wts, candV, candI);
  k3_final<<<1, 256, 0, stream>>>(candV, candI, psum, values, out);
}